// Model_82008105550530
// MI455X (gfx1250) — compile-verified
//
#include <hip/hip_runtime.h>
#include <stdint.h>

// ---------------- problem constants (match reference) ----------------
#define TLEN   4096          // T
#define BATCH  256           // B
#define HID    10            // hidden per direction
#define CHUNK  64            // timesteps per TDM-staged x tile
#define XROW   (CHUNK + 1)   // padded LDS row stride in dwords (bank-conflict-free)
#define XBUF   (16 * XROW)   // dwords per staged x tile
#define HREG   256           // dwords of h transpose region (16x16)

typedef float        v2f __attribute__((ext_vector_type(2)));
typedef float        v8f __attribute__((ext_vector_type(8)));
typedef unsigned int v4u __attribute__((ext_vector_type(4)));
typedef int          v8i __attribute__((ext_vector_type(8)));
typedef int          v4i __attribute__((ext_vector_type(4)));

// ---------------- fast activations (TRANS pipe) ----------------
__device__ __forceinline__ float fexp2(float x) { return __builtin_amdgcn_exp2f(x); }
__device__ __forceinline__ float frcp (float x) { return __builtin_amdgcn_rcpf(x);  }
__device__ __forceinline__ float fsig (float x) {
  return frcp(1.0f + fexp2(-1.44269504f * x));          // 1/(1+e^-x)
}
__device__ __forceinline__ float ftanh(float x) {
#if __has_builtin(__builtin_amdgcn_tanhf)
  return __builtin_amdgcn_tanhf(x);                     // native V_TANH_F32
#elif __has_builtin(__builtin_amdgcn_tanh_f32)
  return __builtin_amdgcn_tanh_f32(x);
#else
  return 1.0f - 2.0f * frcp(1.0f + fexp2(2.88539008f * x)); // 1 - 2/(e^{2x}+1)
#endif
}

// ---------------- TDM: stage a 16(batch) x CHUNK(time) f32 tile into LDS ----------
// 2D strided tile: tile_dim0 = CHUNK (contiguous t), tile_dim1 = 16 (batch rows,
// stride = TLEN elements). pad_enable pads 1 dword per 64 -> LDS row stride 65.
__device__ __forceinline__ void tdm_load_x(const float* gsrc, uint32_t lds_off) {
  uint64_t ga = (uint64_t)(uintptr_t)gsrc;
  v4u g0;
  g0[0] = 1u;                                  // count=1, user descriptor
  g0[1] = lds_off;                             // lds_addr (bytes)
  g0[2] = (uint32_t)ga;                        // global_addr[31:0]
  g0[3] = (uint32_t)(ga >> 32) | (2u << 30);   // global_addr[56:32] | type=2
  v8i g1;
  g1[0] = (2 << 16) | (1 << 20) | (5 << 22);   // data_size=4B, pad_enable, pad_interval=64dw (pad_amount=1dw)
  g1[1] = (CHUNK & 0xFFFF) << 16;              // tensor_dim0[15:0]
  g1[2] = 16 << 16;                            // tensor_dim0[31:16]=0 | tensor_dim1=16
  g1[3] = CHUNK << 16;                         // tensor_dim1[31:16]=0 | tile_dim0=CHUNK
  g1[4] = 16;                                  // tile_dim1=16, tile_dim2=0
  g1[5] = TLEN;                                // tensor_dim0_stride[31:0] (elements)
  g1[6] = 0;                                   // stride0[47:32] | stride1[15:0]
  g1[7] = 0;
  v4i z4 = (v4i){0, 0, 0, 0};                  // groups 2/3 unused (2D tensor)
  v8i z8 = (v8i){0, 0, 0, 0, 0, 0, 0, 0};      // extra descriptor group (6-arg form)
  __builtin_amdgcn_tensor_load_to_lds(g0, g1, z4, z4, z8, 0);
}

__device__ __forceinline__ int chunk_base_t(int ci, int dir) {
  return dir ? (TLEN - (ci + 1) * CHUNK) : ci * CHUNK;
}

// =====================================================================
// Main kernel: one wave = one (16-batch tile, direction) chain.
// gates.T[gate,batch] = [Whh ; Wih^T ; bias^T](64x12 padded) x [h ; x ; 1](12x16)
// via 4 gate-tiles x 3 K-chunks of V_WMMA_F32_16X16X4_F32, f32 throughout.
// =====================================================================
__global__ __launch_bounds__(32, 1) void lstm_dir_kernel(
    const float* __restrict__ x,                        // [B][T]
    const float* __restrict__ Wih_f, const float* __restrict__ Whh_f,
    const float* __restrict__ bih_f, const float* __restrict__ bhh_f,
    const float* __restrict__ Wih_b, const float* __restrict__ Whh_b,
    const float* __restrict__ bih_b, const float* __restrict__ bhh_b,
    const float* __restrict__ Wout,                     // [1][20]
    float* __restrict__ ypart)                          // [2][B][T] partial dots
{
  // unified LDS: [0,256) = h transpose (16 rows x 16 batch),
  //              [256, 256+2*XBUF) = double-buffered x tiles
  __shared__ float smem[HREG + 2 * XBUF];

  const int lane = threadIdx.x;           // 0..31
  const int hi   = lane >> 4;             // lane half (0/1)
  const int bl   = lane & 15;             // batch-in-tile (N) / gate row (M) role
  const int wg   = blockIdx.x;            // 0..31
  const int dir  = wg & 1;                // 0 fwd, 1 bwd
  const int b0   = (wg >> 1) * 16;        // batch tile base

  const float* Wih = dir ? Wih_b : Wih_f;
  const float* Whh = dir ? Whh_b : Whh_f;
  const float* bih = dir ? bih_b : bih_f;
  const float* bhh = dir ? bhh_b : bhh_f;

  // ---- constant A fragments: A[tile][kk] ; layout: lane<16 K=4kk+v, lane>=16 K=4kk+v+2
  // extended weight: k<10 -> Whh[g][k]; k==10 -> Wih[g]; k==11 -> bih[g]+bhh[g]
  v2f A[4][3];
#pragma unroll
  for (int tl = 0; tl < 4; ++tl) {
#pragma unroll
    for (int kk = 0; kk < 3; ++kk) {
#pragma unroll
      for (int v = 0; v < 2; ++v) {
        int k = 4 * kk + v + 2 * hi;      // 0..11
        int g = tl * HID + bl;            // gate tile row -> original gate index
        float val = 0.0f;
        if (bl < HID) {
          if      (k < 10)  val = Whh[g * HID + k];
          else if (k == 10) val = Wih[g];
          else              val = bih[g] + bhh[g];
        }
        A[tl][kk][v] = val;
      }
    }
  }

  // ---- per-lane Wout constants (rows j = r + 8*hi; zero for padded rows) ----
  float wr[8];
#pragma unroll
  for (int r = 0; r < 8; ++r) {
    int j = r + 8 * hi;
    wr[r] = (j < HID) ? Wout[dir * HID + j] : 0.0f;
  }

  // ---- h0 = 0, c0 = 0 ----
  float cst[8];
#pragma unroll
  for (int r = 0; r < 8; ++r) {
    cst[r] = 0.0f;
    smem[(r + 8 * hi) * 16 + bl] = 0.0f;
  }
  __syncthreads();  // single-wave WG: lowers to waits only

  // ---- prefetch first two x chunks (double buffer) ----
  const int nchunks = TLEN / CHUNK;
  tdm_load_x(x + (size_t)b0 * TLEN + chunk_base_t(0, dir),
             (uint32_t)(uintptr_t)&smem[HREG]);
  tdm_load_x(x + (size_t)b0 * TLEN + chunk_base_t(1, dir),
             (uint32_t)(uintptr_t)&smem[HREG + XBUF]);

  for (int ci = 0; ci < nchunks; ++ci) {
    const int cur   = ci & 1;
    const int xbase = HREG + cur * XBUF + bl * XROW;   // this lane's x row base
    __builtin_amdgcn_s_wait_tensorcnt(1);  // current buffer ready (TDM in-order per wave)

    for (int s = 0; s < CHUNK; ++s) {
      const int k    = ci * CHUNK + s;                     // scan index
      const int t    = dir ? (TLEN - 1 - k) : k;           // real timestep
      const int tloc = dir ? (CHUNK - 1 - s) : s;          // position in staged tile

      // ---- build B = [h ; x ; 1] fragments (K=12), fully branchless ----
      v2f Bv[3];
#pragma unroll
      for (int kk = 0; kk < 2; ++kk) {
#pragma unroll
        for (int v = 0; v < 2; ++v) {
          // rows j = 4kk+v+2hi < 10 always: plain h read, arithmetic index
          Bv[kk][v] = smem[(4 * kk + v) * 16 + 32 * hi + bl];
        }
      }
      {
        // kk=2 v0: low half -> h row 8 ; high half -> x[t][b] (index select, 1 ds_load)
        int off = hi ? (xbase + tloc) : (8 * 16 + bl);
        Bv[2][0] = smem[off];
        // kk=2 v1: low half -> h row 9 ; high half -> 1.0 (value select)
        float h9 = smem[9 * 16 + bl];
        Bv[2][1] = hi ? 1.0f : h9;
      }

      // ---- 4 gate tiles x 3 K-chunks of f32 WMMA (chains of 3, 4 independent) ----
      v8f acc[4] = {v8f{}, v8f{}, v8f{}, v8f{}};
#pragma unroll
      for (int kk = 0; kk < 3; ++kk) {
#pragma unroll
        for (int tl = 0; tl < 4; ++tl) {
          acc[tl] = __builtin_amdgcn_wmma_f32_16x16x4_f32(
              false, A[tl][kk], false, Bv[kk], (short)0, acc[tl], false, false);
        }
      }

      // ---- lane-local activations + state update (gate order i,f,g,o) ----
      float ypl = 0.0f;
      float hnew[8];
#pragma unroll
      for (int r = 0; r < 8; ++r) {
        float ig = fsig (acc[0][r]);
        float fg = fsig (acc[1][r]);
        float gg = ftanh(acc[2][r]);
        float og = fsig (acc[3][r]);
        float c  = fg * cst[r] + ig * gg;
        cst[r]   = c;
        float h  = og * ftanh(c);
        hnew[r]  = h;
        ypl      = fmaf(wr[r], h, ypl);     // partial Wout dot (wr=0 on pad rows)
      }

      // ---- publish h for next step's B (LDS transpose; same-wave DS is in-order) ----
#pragma unroll
      for (int r = 0; r < 8; ++r)
        smem[(r + 8 * hi) * 16 + bl] = hnew[r];

      // ---- y partial: combine lane halves, store (only low half writes) ----
      float ysum = ypl + __shfl_xor(ypl, 16, 32);
      if (hi == 0)
        ypart[(size_t)dir * BATCH * TLEN + (size_t)(b0 + bl) * TLEN + t] = ysum;
    }

    // ---- prefetch chunk ci+2 into the buffer we just drained ----
    if (ci + 2 < nchunks)
      tdm_load_x(x + (size_t)b0 * TLEN + chunk_base_t(ci + 2, dir),
                 (uint32_t)(uintptr_t)&smem[HREG + cur * XBUF]);
  }
}

// ---------------- epilogue: y = ypart_fwd + ypart_bwd + bout ----------------
__global__ __launch_bounds__(256) void combine_kernel(
    const float* __restrict__ ws, const float* __restrict__ bout,
    float* __restrict__ out)
{
  size_t i = (size_t)blockIdx.x * 256 + threadIdx.x;   // BATCH*TLEN elements
  out[i] = ws[i] + ws[(size_t)BATCH * TLEN + i] + bout[0];
}

// ---------------- host-side launcher ----------------
extern "C" void kernel_launch(void* const* d_in, const int* in_sizes, int n_in,
                              void* d_out, int out_size, void* d_ws, size_t ws_size,
                              hipStream_t stream) {
  const float* data  = (const float*)d_in[0];   // [B,T,1]
  const float* Wih_f = (const float*)d_in[1];
  const float* Whh_f = (const float*)d_in[2];
  const float* bih_f = (const float*)d_in[3];
  const float* bhh_f = (const float*)d_in[4];
  const float* Wih_b = (const float*)d_in[5];
  const float* Whh_b = (const float*)d_in[6];
  const float* bih_b = (const float*)d_in[7];
  const float* bhh_b = (const float*)d_in[8];
  const float* Wout  = (const float*)d_in[9];
  const float* bout  = (const float*)d_in[10];

  float* ws  = (float*)d_ws;      // needs 2*B*T*4 = 8 MB partial-dot scratch
  float* out = (float*)d_out;     // [B,T,1] f32

  // 32 single-wave workgroups: 16 batch tiles x 2 directions, all concurrent.
  lstm_dir_kernel<<<32, 32, 0, stream>>>(
      data, Wih_f, Whh_f, bih_f, bhh_f,
      Wih_b, Whh_b, bih_b, bhh_b, Wout, ws);

  combine_kernel<<<(BATCH * TLEN) / 256, 256, 0, stream>>>(ws, bout, out);
}